// MyCRF_21277267984643
// MI455X (gfx1250) — compile-verified
//
#include <hip/hip_runtime.h>
#include <hip/hip_bf16.h>

#define B_SZ 512
#define S_SZ 2048
#define L_SZ 17
#define LPAD 20

typedef float v2f __attribute__((ext_vector_type(2)));
typedef float v8f __attribute__((ext_vector_type(8)));

// ---------------------------------------------------------------------------
// Kernel 1: Viterbi forward + backtrace. One wave (32 threads) per batch.
// Lane j owns state j; A-column in 17 VGPRs; alpha broadcast via LDS;
// backpointers kept in LDS (2047 x 20 bytes) so the serial backtrace is
// LDS-latency bound, not HBM-latency bound.
// ---------------------------------------------------------------------------
__global__ void __launch_bounds__(32)
crf_viterbi_kernel(const float* __restrict__ x, const float* __restrict__ A,
                   int* __restrict__ preds) {
    __shared__ float sh_alpha[LPAD];
    __shared__ unsigned char sh_bp[(S_SZ - 1) * LPAD];

    const int lane = threadIdx.x;
    const int b = blockIdx.x;
    const int j = (lane < L_SZ) ? lane : 0;      // lanes 17..31 mirror state 0
    const bool active = (lane < L_SZ);

    // A column for state j: Acol[i] = A[i][j]
    float Acol[L_SZ];
#pragma unroll
    for (int i = 0; i < L_SZ; ++i) Acol[i] = A[i * L_SZ + j];

    const float* xb = x + (size_t)b * S_SZ * L_SZ;

    if (active) sh_alpha[lane] = xb[j];          // alpha_0 = x[b,0,:]
    __syncthreads();

    float xt_next = xb[L_SZ + j];                // t = 1 emission
    for (int t = 1; t < S_SZ; ++t) {
        const float xt = xt_next;
        if (t + 1 < S_SZ) xt_next = xb[(size_t)(t + 1) * L_SZ + j];
        if (((t & 7) == 0) && (t + 8 < S_SZ))
            __builtin_prefetch(xb + (size_t)(t + 8) * L_SZ + j, 0, 0);

        // best over previous states i of alpha[i] + A[i][j]
        float best = sh_alpha[0] + Acol[0];
        int bi = 0;
#pragma unroll
        for (int i = 1; i < L_SZ; ++i) {
            const float sc = sh_alpha[i] + Acol[i];
            if (sc > best) { best = sc; bi = i; }   // strict > == first-argmax
        }
        __syncthreads();
        if (active) {
            sh_bp[(t - 1) * LPAD + lane] = (unsigned char)bi;
            sh_alpha[lane] = best + xt;
        }
        __syncthreads();
    }

    // Backtrace (serial, LDS-resident backpointers)
    if (lane == 0) {
        float bv = sh_alpha[0];
        int lab = 0;
        for (int i = 1; i < L_SZ; ++i)
            if (sh_alpha[i] > bv) { bv = sh_alpha[i]; lab = i; }
        int* pb = preds + (size_t)b * S_SZ;
        pb[S_SZ - 1] = lab;
        for (int t = S_SZ - 1; t >= 1; --t) {
            lab = (int)sh_bp[(t - 1) * LPAD + lab];
            pb[t - 1] = lab;
        }
    }
}

// ---------------------------------------------------------------------------
// Kernel 2: log-partition scan via fp32 WMMA.
// One wave handles 16 batches (rows). Per step:
//   u = alpha + x_t ; m = max(u) ; v = exp(u - m)            (registers)
//   alpha'[0:16] = m + log( (V x expA)[j] )   via 5x V_WMMA_F32_16X16X4_F32
//   alpha'[16]   = m + log( dot(v, expA[:,16]) )             (17 scalar FMAs)
// expA B-fragments are constant and preloaded in registers.
// Lanes l and l+16 both own row (l&15) so A-fragment K-halves come straight
// from registers (branchless cndmask select) -- no LDS needed on the A side.
// D (8 VGPRs, column-striped) is transposed back to row-per-lane via LDS.
// ---------------------------------------------------------------------------
__global__ void __launch_bounds__(32)
crf_partition_kernel(const float* __restrict__ x, const float* __restrict__ A,
                     float* __restrict__ Zout) {
    __shared__ float shD[16 * LPAD];

    const int lane = threadIdx.x;
    const int col = lane & 15;        // D column / B column owned by this lane
    const int half = lane >> 4;       // K-half for A/B fragments
    const int row = col;              // batch row mirrored by this lane
    const int b = blockIdx.x * 16 + row;

    // Constant B fragments: bf[kb] holds expA rows K=4kb+2*half, +1 (cols = col)
    v2f bf[5];
#pragma unroll
    for (int kb = 0; kb < 5; ++kb) {
        const int k0 = 4 * kb + 2 * half;
        bf[kb].x = (k0 < L_SZ) ? __expf(A[k0 * L_SZ + col]) : 0.0f;
        bf[kb].y = (k0 + 1 < L_SZ) ? __expf(A[(k0 + 1) * L_SZ + col]) : 0.0f;
    }
    // expA column 16 (N-tile remainder handled scalar)
    float eac16[L_SZ];
#pragma unroll
    for (int i = 0; i < L_SZ; ++i) eac16[i] = __expf(A[i * L_SZ + 16]);

    const float* xb = x + (size_t)b * S_SZ * L_SZ;

    float alpha[L_SZ];
#pragma unroll
    for (int i = 0; i < L_SZ; ++i) alpha[i] = 0.0f;

    float xc[L_SZ], xn[L_SZ];
#pragma unroll
    for (int i = 0; i < L_SZ; ++i) xc[i] = xb[i];

    for (int t = 0; t < S_SZ - 1; ++t) {
        // double-buffer next emission row to hide L2 latency
        const float* xr = xb + (size_t)(t + 1) * L_SZ;
#pragma unroll
        for (int i = 0; i < L_SZ; ++i) xn[i] = xr[i];

        float u[L_SZ];
        float m = -3.4e38f;
#pragma unroll
        for (int i = 0; i < L_SZ; ++i) {
            u[i] = alpha[i] + xc[i];
            m = fmaxf(m, u[i]);
        }
        float v[20];
#pragma unroll
        for (int i = 0; i < L_SZ; ++i) v[i] = __expf(u[i] - m);
        v[17] = 0.0f; v[18] = 0.0f; v[19] = 0.0f;

        float s16 = 0.0f;
#pragma unroll
        for (int i = 0; i < L_SZ; ++i) s16 = fmaf(v[i], eac16[i], s16);

        v8f acc = {0.f, 0.f, 0.f, 0.f, 0.f, 0.f, 0.f, 0.f};
#pragma unroll
        for (int kb = 0; kb < 5; ++kb) {
            const float ax0 = v[4 * kb + 0];
            const float ay0 = v[4 * kb + 1];
            const float ax1 = v[4 * kb + 2];
            const float ay1 = v[4 * kb + 3];
            v2f a;
            a.x = half ? ax1 : ax0;
            a.y = half ? ay1 : ay0;
            acc = __builtin_amdgcn_wmma_f32_16x16x4_f32(
                false, a, false, bf[kb], (short)0, acc, false, false);
        }

        __syncthreads();   // previous-iteration reads of shD are done
#pragma unroll
        for (int r = 0; r < 8; ++r)
            shD[(r + 8 * half) * LPAD + col] = acc[r];
        __syncthreads();

#pragma unroll
        for (int jj = 0; jj < 16; ++jj)
            alpha[jj] = m + __logf(shD[row * LPAD + jj]);
        alpha[16] = m + __logf(s16);

#pragma unroll
        for (int i = 0; i < L_SZ; ++i) xc[i] = xn[i];
    }

    // Final combine: Z = logsumexp(alpha + x[:, S-1, :])
    float m = -3.4e38f;
    float u[L_SZ];
#pragma unroll
    for (int i = 0; i < L_SZ; ++i) {
        u[i] = alpha[i] + xc[i];
        m = fmaxf(m, u[i]);
    }
    float s = 0.0f;
#pragma unroll
    for (int i = 0; i < L_SZ; ++i) s += __expf(u[i] - m);
    if (half == 0) Zout[b] = m + __logf(s);
}

// ---------------------------------------------------------------------------
// Kernel 3: per-batch emission + transition score (gold path). x is in L2
// after the scans, so the 17-wide gathers are cheap.
// ---------------------------------------------------------------------------
__global__ void __launch_bounds__(256)
crf_emistrans_kernel(const float* __restrict__ x, const long long* __restrict__ y,
                     const float* __restrict__ A, float* __restrict__ et) {
    __shared__ float red[256];
    const int b = blockIdx.x;
    const int tid = threadIdx.x;
    const long long* yb = y + (size_t)b * S_SZ;
    const float* xb = x + (size_t)b * S_SZ * L_SZ;

    float acc = 0.0f;
    for (int t = tid; t < S_SZ; t += 256) {
        const int yt = (int)yb[t];
        acc += xb[(size_t)t * L_SZ + yt];
        if (t < S_SZ - 1) {
            const int yn = (int)yb[t + 1];
            acc += A[yn * L_SZ + yt];     // A[y_{t+1}, y_t]
        }
    }
    red[tid] = acc;
    __syncthreads();
    for (int s = 128; s > 0; s >>= 1) {
        if (tid < s) red[tid] += red[tid + s];
        __syncthreads();
    }
    if (tid == 0) et[b] = red[0];
}

// ---------------------------------------------------------------------------
// Kernel 4: nll = mean_b( Z_b - (emis_b + trans_b) )
// ---------------------------------------------------------------------------
__global__ void __launch_bounds__(512)
crf_nll_kernel(const float* __restrict__ Z, const float* __restrict__ et,
               float* __restrict__ out) {
    __shared__ float red[512];
    const int tid = threadIdx.x;
    red[tid] = Z[tid] - et[tid];
    __syncthreads();
    for (int s = 256; s > 0; s >>= 1) {
        if (tid < s) red[tid] += red[tid + s];
        __syncthreads();
    }
    if (tid == 0) out[0] = red[0] / (float)B_SZ;
}

// ---------------------------------------------------------------------------
extern "C" void kernel_launch(void* const* d_in, const int* in_sizes, int n_in,
                              void* d_out, int out_size, void* d_ws, size_t ws_size,
                              hipStream_t stream) {
    const float* x = (const float*)d_in[0];            // (512, 2048, 17) f32
    const long long* y = (const long long*)d_in[1];    // (512, 2048) i64
    const float* A = (const float*)d_in[2];            // (17, 17) f32

    int* preds = (int*)d_out;                                  // first B*S outputs
    float* nll = (float*)d_out + (size_t)B_SZ * S_SZ;          // final scalar

    float* wsZ = (float*)d_ws;          // 512 floats: log-partition per batch
    float* wsET = wsZ + B_SZ;           // 512 floats: emis+trans per batch

    crf_viterbi_kernel<<<B_SZ, 32, 0, stream>>>(x, A, preds);
    crf_partition_kernel<<<B_SZ / 16, 32, 0, stream>>>(x, A, wsZ);
    crf_emistrans_kernel<<<B_SZ, 256, 0, stream>>>(x, y, A, wsET);
    crf_nll_kernel<<<1, B_SZ, 0, stream>>>(wsZ, wsET, nll);
}